// SpatialWeight_5858335392304
// MI455X (gfx1250) — compile-verified
//
#include <hip/hip_runtime.h>

typedef __attribute__((ext_vector_type(2))) float v2f;
typedef __attribute__((ext_vector_type(4))) float v4f;
typedef __attribute__((ext_vector_type(8))) float v8f;

#define HW    65536          // 256*256 pixels per (b,f) image
#define NCH   64             // channels
#define NBF   16             // b*f = 2*8
#define NSUM  (NBF * NCH)    // 1024 channel-plane sums

// ---------------------------------------------------------------------------
// Phase 1: ksum[bf*64 + c] = sum over all 65536 pixels of nbr[bf, c, :, :]
// One block per channel plane (256 KB contiguous). Deterministic tree reduce.
// Non-temporal loads: 256 MB streamed once, keep it out of L2 retention.
// ---------------------------------------------------------------------------
__global__ __launch_bounds__(256) void ksum_kernel(const float* __restrict__ nbr,
                                                   float* __restrict__ ksum) {
    const int ch = blockIdx.x;  // 0..1023
    const v4f* p = (const v4f*)nbr + (size_t)ch * (HW / 4) + threadIdx.x;

    float s = 0.f;
#pragma unroll 8
    for (int i = 0; i < (HW / 4) / 256; ++i) {
        v4f v = __builtin_nontemporal_load(p + (size_t)i * 256);
        s += (v.x + v.y) + (v.z + v.w);
    }

    __shared__ float red[256];
    red[threadIdx.x] = s;
    __syncthreads();
    for (int off = 128; off > 0; off >>= 1) {
        if ((int)threadIdx.x < off) red[threadIdx.x] += red[threadIdx.x + off];
        __syncthreads();
    }
    if (threadIdx.x == 0) ksum[ch] = red[0];
}

// ---------------------------------------------------------------------------
// Phase 2: out[bf, p] = sigmoid( sum_c ref[bf, c, p] * ksum[bf, c] )
// Per wave: 4 tiles of 16 pixels. Each tile = 16 chained V_WMMA_F32_16X16X4_F32:
//   A[m,k] = ksum[4i+k] (replicated over rows m)   -> 2 VGPRs/lane
//   B[k,n] = ref[4i+k, pix+n]                      -> 2 VGPRs/lane
//   D[m,n] accumulates dot(ref[:,pix+n], ksum)  (all rows identical)
// VGPR layouts per CDNA5 ISA 7.12.2: lanes 0-15 supply K=0,1; lanes 16-31 K=2,3.
// ---------------------------------------------------------------------------
#define TILES 4

__global__ __launch_bounds__(256) void weight_kernel(const float* __restrict__ ref,
                                                     const float* __restrict__ ksum,
                                                     float* __restrict__ out) {
    const int lane  = threadIdx.x & 31;
    const int wave  = threadIdx.x >> 5;       // 0..7
    const int half2 = (lane >> 4) << 1;       // 0 for lanes 0-15, 2 for lanes 16-31
    const int n     = lane & 15;              // N column / pixel within tile

    const unsigned group = blockIdx.x * 8u + wave;  // 64 pixels per group, 16384 groups
    const unsigned bf    = group >> 10;             // 1024 groups per (b,f) image
    const unsigned pix0  = (group & 1023u) * 64u;   // first pixel within image

    const float* __restrict__ ks   = ksum + bf * NCH;
    const float* __restrict__ rimg = ref + (size_t)bf * NCH * HW;

    // Preload A-matrix registers: ksum values this lane supplies (32 floats, 4 KB table -> cached)
    float a0[16], a1[16];
#pragma unroll
    for (int i = 0; i < 16; ++i) {
        a0[i] = ks[4 * i + half2];
        a1[i] = ks[4 * i + half2 + 1];
    }

    for (int t = 0; t < TILES; ++t) {
        const unsigned pix = pix0 + t * 16;
        const float* bp = rimg + (size_t)half2 * HW + pix + n;

        v8f acc = {0.f, 0.f, 0.f, 0.f, 0.f, 0.f, 0.f, 0.f};
#pragma unroll
        for (int i = 0; i < 16; ++i) {
            v2f A;
            A.x = a0[i];
            A.y = a1[i];
            v2f B;
            B.x = __builtin_nontemporal_load(bp + (size_t)(4 * i) * HW);
            B.y = __builtin_nontemporal_load(bp + (size_t)(4 * i + 1) * HW);
            // 8 args: (neg_a, A, neg_b, B, c_mod, C, reuse_a, reuse_b)
            acc = __builtin_amdgcn_wmma_f32_16x16x4_f32(
                false, A, false, B, (short)0, acc, false, false);
        }

        // D: every row equals the dot product for pixel pix+n; lane's VGPR0 holds it.
        if (lane < 16) {
            float x = acc[0];
            // Fast sigmoid: v_exp_f32 + v_add_f32 + v_rcp_f32 (skip IEEE div sequence)
            float e   = __expf(-x);
            float sgm = __builtin_amdgcn_rcpf(1.f + e);
            __builtin_nontemporal_store(sgm, out + (size_t)bf * HW + pix + n);
        }
    }
}

// ---------------------------------------------------------------------------
extern "C" void kernel_launch(void* const* d_in, const int* in_sizes, int n_in,
                              void* d_out, int out_size, void* d_ws, size_t ws_size,
                              hipStream_t stream) {
    const float* nbr = (const float*)d_in[0];   // [2,8,64,256,256] f32
    const float* ref = (const float*)d_in[1];   // [2,8,64,256,256] f32
    float* out  = (float*)d_out;                // [2,8,1,256,256]  f32
    float* ksum = (float*)d_ws;                 // 1024 floats of scratch

    // Phase 1: 1024 channel-plane reductions
    ksum_kernel<<<NSUM, 256, 0, stream>>>(nbr, ksum);

    // Phase 2: 16384 pixel-groups of 64 pixels, 8 waves per block
    weight_kernel<<<16384 / 8, 256, 0, stream>>>(ref, ksum, out);
}